// SG_EnvmapMaterialNet_12953621364845
// MI455X (gfx1250) — compile-verified
//
#include <hip/hip_runtime.h>
#include <stdint.h>

#define TINY_F      1e-6f
#define INV_PI_F    0.31830988618379067154f
#define TWO_PI_F    6.28318530717958647692f
#define MU_COS_F    32.708f
#define LAMBDA_COS_F 0.0315f
#define ALPHA_COS_F 31.7003f
#define M_LOBES     128

typedef unsigned int v4u __attribute__((ext_vector_type(4)));
typedef int          v4i __attribute__((ext_vector_type(4)));
typedef int          v8i __attribute__((ext_vector_type(8)));

__device__ __forceinline__ float fexp(float x) {
  // e^x = 2^(x*log2 e) -> single v_exp_f32
  return __builtin_amdgcn_exp2f(x * 1.44269504088896340736f);
}
__device__ __forceinline__ float frcp(float x)  { return __builtin_amdgcn_rcpf(x); }
__device__ __forceinline__ float fsqrtf_(float x){ return __builtin_amdgcn_sqrtf(x); }

// hemisphere_int with lam-dependent part precomputed (t, A=e^-t, A_b, A_u).
// cos>=0 -> inv_b=e^{-t*cos}, b=1 ; cos<0 -> inv_b=1, b=e^{t*cos}.
// Both equal u = e^{-t*|cos|}, so one exp + select.
__device__ __forceinline__ float hemi_sel(float t, float A, float Ab, float Au, float c) {
  float u  = fexp(-t * fabsf(c));
  float s1 = (1.0f - A * u) * frcp((1.0f - A) + u - A * u);
  float s2 = (u - A) * frcp((1.0f - A) * (u + 1.0f));
  float s  = (c >= 0.0f) ? s1 : s2;
  return Ab + (Au - Ab) * s;   // Ab*(1-s) + Au*s
}

__device__ __forceinline__ float hemi(float lam, float c) {
  float lamt = lam + TINY_F;
  float inv  = frcp(lamt);
  float t    = fsqrtf_(lamt) * (1.6988f + 10.8438f * inv) *
               frcp(1.0f + 6.2201f * inv + 10.2415f * inv * inv);
  float A    = fexp(-t);
  float ea   = fexp(-lamt);
  float Ab   = TWO_PI_F * inv * (ea - ea * ea);   // e^-lam - e^-2lam
  float Au   = TWO_PI_F * inv * (1.0f - ea);
  return hemi_sel(t, A, Ab, Au, c);
}

__global__ __launch_bounds__(256) void sg_envmap_kernel(
    const float* __restrict__ lgt,     // (128,7)
    const float* __restrict__ specR,   // (1,3)
    const float* __restrict__ rough,   // (1,1)
    const float* __restrict__ albedo,  // (N,3)
    const float* __restrict__ nrmG,    // (N,3)
    const float* __restrict__ vdirG,   // (N,3)
    const float* __restrict__ bw,      // (N,1)
    float* __restrict__ out,           // (N,3)
    int N)
{
  __shared__ float  sRaw[M_LOBES * 7];          // staged lgtSGs
  __shared__ float4 sSG[M_LOBES * 3];           // per-lobe precomputed rows

  // ---- Stage lgtSGs (3584 B) into LDS via CDNA5 Tensor Data Mover ----------
#if __has_builtin(__builtin_amdgcn_tensor_load_to_lds) && __has_builtin(__builtin_amdgcn_s_wait_tensorcnt)
  if (threadIdx.x < 32) {   // one wave issues the TDM op (EXEC-independent)
    unsigned ldsOff = (unsigned)((uintptr_t)(void*)&sRaw[0] & 0xFFFFFFFFu);
    unsigned long long ga = (unsigned long long)(uintptr_t)lgt;
    v4u g0;
    g0[0] = 1u;                                              // count=1, user mode
    g0[1] = ldsOff;                                          // lds_addr
    g0[2] = (unsigned)(ga & 0xFFFFFFFFull);                  // global_addr[31:0]
    g0[3] = (unsigned)((ga >> 32) & 0x01FFFFFFull)           // global_addr[56:32]
          | 0x80000000u;                                     // type=2 ("image")
    v8i g1;
    g1[0] = (int)(2u << 16);      // workgroup_mask=0 (no cluster), data_size=2 (4B)
    g1[1] = (int)(896u << 16);    // tensor_dim0 = 896 (bits 79:48, low 16 here)
    g1[2] = (int)(1u << 16);      // tensor_dim0 hi = 0 | tensor_dim1 = 1 (low 16)
    g1[3] = (int)(896u << 16);    // tensor_dim1 hi = 0 | tile_dim0 = 896
    g1[4] = 1;                    // tile_dim1 = 1, tile_dim2 = 0 (unused)
    g1[5] = 896;                  // tensor_dim0_stride low 32
    g1[6] = (int)(896u << 16);    // stride0 hi = 0 | tensor_dim1_stride low 16
    g1[7] = 0;                    // stride1 hi
    v4i g2 = {0, 0, 0, 0};        // dims 2/3 unused
    v4i g3 = {0, 0, 0, 0};
    v8i gz = {0, 0, 0, 0, 0, 0, 0, 0};   // extra group (clang-23 6-arg form), unused
    __builtin_amdgcn_tensor_load_to_lds(g0, g1, g2, g3, gz, 0);
    __builtin_amdgcn_s_wait_tensorcnt(0);                    // TENSORcnt -> 0
  }
  __syncthreads();
#endif
  // Belt-and-braces: redundantly copy the same 896 floats. If the TDM D#
  // descriptor was interpreted correctly this rewrites identical values
  // (L2-resident, ~3.5 KB per block, negligible); if not, it fixes sRaw.
  // sSG is only written after the next barrier, so even a misdirected TDM
  // (OOB LDS writes are dropped in HW) cannot corrupt the final state.
  for (int i = threadIdx.x; i < M_LOBES * 7; i += blockDim.x) sRaw[i] = lgt[i];
  __syncthreads();

  // ---- Per-lobe precompute, amortized over all pixels in the block ---------
  {
    int t = threadIdx.x;
    if (t < M_LOBES) {
      const float* r = &sRaw[t * 7];
      float lx = r[0], ly = r[1], lz = r[2];
      float lam = fabsf(r[3]);
      float m0 = fabsf(r[4]), m1 = fabsf(r[5]), m2 = fabsf(r[6]);
      float inn = frcp(fsqrtf_(lx * lx + ly * ly + lz * lz) + TINY_F);
      lx *= inn; ly *= inn; lz *= inn;
      float ratio_d = LAMBDA_COS_F * frcp(lam);            // lam1/lam2 (diffuse)
      // hemisphere_int(lgtLam, .) lam-only terms:
      float lamt = lam + TINY_F;
      float inv  = frcp(lamt);
      float tt   = fsqrtf_(lamt) * (1.6988f + 10.8438f * inv) *
                   frcp(1.0f + 6.2201f * inv + 10.2415f * inv * inv);
      float A    = fexp(-tt);
      float ea   = fexp(-lamt);
      float Ab   = TWO_PI_F * inv * (ea - ea * ea);
      float Au   = TWO_PI_F * inv * (1.0f - ea);
      sSG[t * 3 + 0] = make_float4(lx, ly, lz, lam);
      sSG[t * 3 + 1] = make_float4(m0, m1, m2, ratio_d);
      sSG[t * 3 + 2] = make_float4(tt, A, Ab, Au);
    }
  }
  __syncthreads();

  int n = blockIdx.x * blockDim.x + threadIdx.x;
  if (n >= N) return;

  // ---- Per-pixel precompute (independent of m) -----------------------------
  float nx = nrmG[3 * n], ny = nrmG[3 * n + 1], nz = nrmG[3 * n + 2];
  float vx = vdirG[3 * n], vy = vdirG[3 * n + 1], vz = vdirG[3 * n + 2];
  float a0 = albedo[3 * n] * INV_PI_F;
  float a1 = albedo[3 * n + 1] * INV_PI_F;
  float a2 = albedo[3 * n + 2] * INV_PI_F;
  float wgt = bw[n];                       // K == 1

  float r0 = rough[0];
  float inv_r4  = frcp(r0 * r0 * r0 * r0);
  float brdfLam = 2.0f * inv_r4;
  float brdfMu  = inv_r4 * INV_PI_F;
  float sRr = specR[0], sRg = specR[1], sRb = specR[2];

  float vdl = fmaxf(nx * vx + ny * vy + nz * vz, 0.0f);     // v_dot_lobe
  float wx = 2.0f * vdl * nx - vx;
  float wy = 2.0f * vdl * ny - vy;
  float wz = 2.0f * vdl * nz - vz;
  float innw = frcp(fsqrtf_(wx * wx + wy * wy + wz * wz) + TINY_F);
  wx *= innw; wy *= innw; wz *= innw;                        // warpLobes
  float warpLam = brdfLam * frcp(4.0f * vdl + TINY_F);

  float hx = wx + vx, hy = wy + vy, hz = wz + vz;
  float innh = frcp(fsqrtf_(hx * hx + hy * hy + hz * hz) + TINY_F);
  hx *= innh; hy *= innh; hz *= innh;                        // new_half
  float vdh = fmaxf(vx * hx + vy * hy + vz * hz, 0.0f);
  float p2  = __builtin_amdgcn_exp2f(-(5.55473f * vdh + 6.8316f) * vdh);
  float F0 = sRr + (1.0f - sRr) * p2;
  float F1 = sRg + (1.0f - sRg) * p2;
  float F2 = sRb + (1.0f - sRb) * p2;

  float dot1 = fmaxf(wx * nx + wy * ny + wz * nz, 0.0f);
  float dot2 = vdl;                                          // clamp(dot(v,n))
  float kk = (r0 + 1.0f) * (r0 + 1.0f) * 0.125f;
  float G1 = dot1 * frcp(dot1 * (1.0f - kk) + kk + TINY_F);
  float G2 = dot2 * frcp(dot2 * (1.0f - kk) + kk + TINY_F);
  float Gs = G1 * G2 * frcp(4.0f * dot1 * dot2 + TINY_F);
  float wm0 = brdfMu * F0 * Gs;                              // warpMus
  float wm1 = brdfMu * F1 * Gs;
  float wm2 = brdfMu * F2 * Gs;
  float invWarpLam = frcp(warpLam);

  float sp0 = 0.0f, sp1 = 0.0f, sp2 = 0.0f;
  float df0 = 0.0f, df1 = 0.0f, df2 = 0.0f;

  // ---- Main loop over 128 SG lobes (LDS broadcast reads) -------------------
#pragma unroll 2
  for (int m = 0; m < M_LOBES; ++m) {
    float4 q0 = sSG[3 * m + 0];
    float4 q1 = sSG[3 * m + 1];
    float4 q2 = sSG[3 * m + 2];
    float lx = q0.x, ly = q0.y, lz = q0.z, lam = q0.w;
    float mu0 = q1.x, mu1 = q1.y, mu2 = q1.z, ratio_d = q1.w;

    // -- specular: lambda_trick(lgt, warp)
    float ratio1 = lam * invWarpLam;
    float dL = lx * wx + ly * wy + lz * wz;
    float tmp1 = fsqrtf_(ratio1 * ratio1 + 1.0f + 2.0f * ratio1 * dL);
    tmp1 = fminf(tmp1, ratio1 + 1.0f);
    float fla = warpLam * tmp1;
    float e1  = fexp(warpLam * (tmp1 - ratio1 - 1.0f));
    float it1 = frcp(tmp1);
    float flx = (ratio1 * lx + wx) * it1;
    float fly = (ratio1 * ly + wy) * it1;
    float flz = (ratio1 * lz + wz) * it1;
    float fmu0 = mu0 * wm0 * e1;
    float fmu1 = mu1 * wm1 * e1;
    float fmu2 = mu2 * wm2 * e1;

    // -- specular: lambda_trick(n/LAMBDA_COS/MU_COS, fl)
    float ratio2 = LAMBDA_COS_F * frcp(fla);
    float dN = nx * flx + ny * fly + nz * flz;               // also d2
    float tmp2 = fsqrtf_(ratio2 * ratio2 + 1.0f + 2.0f * ratio2 * dN);
    tmp2 = fminf(tmp2, ratio2 + 1.0f);
    float lap = fla * tmp2;
    float e2  = MU_COS_F * fexp(fla * (tmp2 - ratio2 - 1.0f));
    float it2 = frcp(tmp2);
    float lpx = (ratio2 * nx + flx) * it2;
    float lpy = (ratio2 * ny + fly) * it2;
    float lpz = (ratio2 * nz + flz) * it2;
    float d1 = lpx * nx + lpy * ny + lpz * nz;
    float h1 = hemi(lap, d1);
    float h2 = ALPHA_COS_F * hemi(fla, dN);
    float gsp = e2 * h1 - h2;              // common scalar factor
    sp0 += fmu0 * gsp;
    sp1 += fmu1 * gsp;
    sp2 += fmu2 * gsp;

    // -- diffuse: lambda_trick(n/LAMBDA_COS/MU_COS, lgt lobe)
    float fd0 = mu0 * a0, fd1 = mu1 * a1, fd2 = mu2 * a2;    // lgtMu * albedo/pi
    float dNL = nx * lx + ny * ly + nz * lz;                 // also dd2
    float tmp3 = fsqrtf_(ratio_d * ratio_d + 1.0f + 2.0f * ratio_d * dNL);
    tmp3 = fminf(tmp3, ratio_d + 1.0f);
    float lap2 = lam * tmp3;
    float e3   = MU_COS_F * fexp(lam * (tmp3 - ratio_d - 1.0f));
    float it3  = frcp(tmp3);
    float qx = (ratio_d * nx + lx) * it3;
    float qy = (ratio_d * ny + ly) * it3;
    float qz = (ratio_d * nz + lz) * it3;
    float dd1 = qx * nx + qy * ny + qz * nz;
    float h3 = hemi(lap2, dd1);
    float h4 = ALPHA_COS_F * hemi_sel(q2.x, q2.y, q2.z, q2.w, dNL);  // precomputed lam part
    float gdf = e3 * h3 - h4;
    df0 += fd0 * gdf;
    df1 += fd1 * gdf;
    df2 += fd2 * gdf;
  }

  float o0 = fmaxf(sp0 * wgt, 0.0f) + fmaxf(df0, 0.0f);
  float o1 = fmaxf(sp1 * wgt, 0.0f) + fmaxf(df1, 0.0f);
  float o2 = fmaxf(sp2 * wgt, 0.0f) + fmaxf(df2, 0.0f);
  out[3 * n + 0] = o0;
  out[3 * n + 1] = o1;
  out[3 * n + 2] = o2;
}

extern "C" void kernel_launch(void* const* d_in, const int* in_sizes, int n_in,
                              void* d_out, int out_size, void* d_ws, size_t ws_size,
                              hipStream_t stream) {
  (void)n_in; (void)out_size; (void)d_ws; (void)ws_size;
  const float* lgt    = (const float*)d_in[0];  // (128,7)
  const float* specR  = (const float*)d_in[1];  // (1,3)
  const float* rough  = (const float*)d_in[2];  // (1,1)
  const float* albedo = (const float*)d_in[3];  // (N,3)
  const float* nrm    = (const float*)d_in[4];  // (N,3)
  const float* vdir   = (const float*)d_in[5];  // (N,3)
  const float* bw     = (const float*)d_in[6];  // (N,1)
  float* out = (float*)d_out;

  int N = in_sizes[4] / 3;                      // normal is (N,3)
  int blocks = (N + 255) / 256;
  sg_envmap_kernel<<<blocks, 256, 0, stream>>>(lgt, specR, rough, albedo, nrm,
                                               vdir, bw, out, N);
}